// GraphSSM_11699490914490
// MI455X (gfx1250) — compile-verified
//
#include <hip/hip_runtime.h>

// Problem constants (match the reference).
#define T_STEPS 32
#define DIM     256
#define D_STATE 16

typedef __attribute__((ext_vector_type(2))) float v2f;
typedef __attribute__((ext_vector_type(4))) float v4f;
typedef __attribute__((ext_vector_type(8))) float v8f;

// ---------------------------------------------------------------------------
// Kernel 1: row-means of A[256,16] and B[256,16] via V_WMMA_F32_16X16X4_F32.
//   D(16x16) = Atile(16x4) x Ones(4x16)/16, accumulated over K = 0..15.
//   Every column of D then holds the row means; we read column N=0
//   (lane 0 -> rows m0..m0+7, lane 16 -> rows m0+8..m0+15 per C/D layout).
// Launch: grid=2 (block 0 -> A, block 1 -> B), block=32 (one wave, EXEC all 1s).
// ---------------------------------------------------------------------------
__global__ __launch_bounds__(32) void rowmean16_wmma_kernel(
    const float* __restrict__ A, const float* __restrict__ B,
    float* __restrict__ abar, float* __restrict__ bbar) {
  const float* __restrict__ src = (blockIdx.x == 0) ? A : B;
  float* __restrict__ dst       = (blockIdx.x == 0) ? abar : bbar;

  const int lane  = threadIdx.x;        // 0..31
  const int row   = lane & 15;          // M index within tile
  const int khalf = (lane >> 4) & 1;    // lanes 16..31 carry K+2, K+3

  v2f ones;                             // B operand: all entries 1/16 (uniform,
  ones.x = 0.0625f;                     // so the 4x16 VGPR layout is moot)
  ones.y = 0.0625f;

  for (int tile = 0; tile < DIM / 16; ++tile) {
    const int m0 = tile * 16;
    v8f c = {};
#pragma unroll
    for (int k0 = 0; k0 < D_STATE; k0 += 4) {
      v2f a;  // A-matrix 16x4 f32 layout: VGPR0=K, VGPR1=K+1 (lanes 0-15),
              //                           VGPR0=K+2, VGPR1=K+3 (lanes 16-31)
      a.x = src[(m0 + row) * D_STATE + k0 + 2 * khalf + 0];
      a.y = src[(m0 + row) * D_STATE + k0 + 2 * khalf + 1];
      c = __builtin_amdgcn_wmma_f32_16x16x4_f32(
          /*neg_a=*/false, a, /*neg_b=*/false, ones,
          /*c_mod=*/(short)0, c, /*reuse_a=*/false, /*reuse_b=*/false);
    }
    // Column N=0 of D lives in lanes 0 (M = m0..m0+7) and 16 (M = m0+8..m0+15).
    if ((lane & 15) == 0) {
      const int base = m0 + (khalf ? 8 : 0);
#pragma unroll
      for (int v = 0; v < 8; ++v) dst[base + v] = c[v];
    }
  }
}

// ---------------------------------------------------------------------------
// Kernel 2: memory-bound streaming scan.
//   out[t,e] = Abar[e%DIM] * out[t-1,e] + Bbar[e%DIM] * x[t,e]
// One thread owns 4 consecutive channels of one node across all 32 timesteps:
// fully coalesced b128 NT loads/stores (655 MB total -> ~28 us @ 23.3 TB/s).
// ---------------------------------------------------------------------------
__global__ __launch_bounds__(256) void ssm_scan_kernel(
    const float* __restrict__ seq, const float* __restrict__ abar,
    const float* __restrict__ bbar, float* __restrict__ out,
    long long total4 /* (N*DIM)/4 */, long long stride /* N*DIM */) {
  __shared__ float sA[DIM];
  __shared__ float sB[DIM];
  for (int i = threadIdx.x; i < DIM; i += blockDim.x) {
    sA[i] = abar[i];
    sB[i] = bbar[i];
  }
  __syncthreads();

  const long long idx4 = (long long)blockIdx.x * blockDim.x + threadIdx.x;
  if (idx4 >= total4) return;

  const long long off = idx4 * 4;
  const int dbase = (int)(off & (DIM - 1));   // DIM divides 4-aligned offsets
  const v4f a4 = *(const v4f*)(&sA[dbase]);   // ds_load_b128
  const v4f b4 = *(const v4f*)(&sB[dbase]);

  v4f h = {0.0f, 0.0f, 0.0f, 0.0f};
#pragma unroll 4
  for (int t = 0; t < T_STEPS; ++t) {
    const v4f x =
        __builtin_nontemporal_load((const v4f*)(seq + off + (long long)t * stride));
    h.x = fmaf(a4.x, h.x, b4.x * x.x);
    h.y = fmaf(a4.y, h.y, b4.y * x.y);
    h.z = fmaf(a4.z, h.z, b4.z * x.z);
    h.w = fmaf(a4.w, h.w, b4.w * x.w);
    __builtin_nontemporal_store(h, (v4f*)(out + off + (long long)t * stride));
  }
}

extern "C" void kernel_launch(void* const* d_in, const int* in_sizes, int n_in,
                              void* d_out, int out_size, void* d_ws, size_t ws_size,
                              hipStream_t stream) {
  const float* seq = (const float*)d_in[0];   // [T, N, DIM] f32
  const float* A   = (const float*)d_in[1];   // [DIM, D_STATE] f32
  const float* B   = (const float*)d_in[2];   // [DIM, D_STATE] f32
  float* out       = (float*)d_out;           // [T, N, DIM] f32

  float* abar = (float*)d_ws;                 // [DIM]
  float* bbar = abar + DIM;                   // [DIM]

  // Stage 1: coefficient means via WMMA (2 waves total).
  rowmean16_wmma_kernel<<<2, 32, 0, stream>>>(A, B, abar, bbar);

  // Stage 2: streaming scan over all (n, d) channels.
  const long long total  = (long long)in_sizes[0];       // T*N*DIM
  const long long stride = total / T_STEPS;              // N*DIM
  const long long total4 = stride / 4;
  const int blocks = (int)((total4 + 255) / 256);
  ssm_scan_kernel<<<blocks, 256, 0, stream>>>(seq, abar, bbar, out, total4, stride);
}